// ReEig_9423158247635
// MI455X (gfx1250) — compile-verified
//
#include <hip/hip_runtime.h>

// ReEig via matrix-sign Newton-Schulz: out = 0.5*(X + EP*I + |X - EP*I|)
// |M| = s * (M/s) * sign(M),  sign via Y <- 1.5Y - 0.5*Y^3  (all 64x64 f32 GEMMs
// on V_WMMA_F32_16X16X4_F32). One workgroup (8 wave32) per batch matrix.
//
// All operands (A0, Y, T=Y^2) are symmetric (polynomials of the same symmetric
// matrix), so B-fragments are loaded along ROWS (contiguous float2) instead of
// columns -> every LDS fragment read is an aligned b64, halving DS traffic.

typedef __attribute__((ext_vector_type(2))) float v2f;
typedef __attribute__((ext_vector_type(8))) float v8f;

#define PITCH    66          // 64 + 2: even pitch -> 8B-aligned float2 LDS loads
#define NS_ITERS 18
#define EPV      1e-4f

// Two 16x16 output tiles (same row block ti, columns tj0/tj1) of a 64x64x64
// symmetric GEMM, accumulated with WMMA. A fragment shared by both tiles.
// Fragment layout: lane&15 = M (or N), lane>>4 picks K pair, 2 VGPRs = K in pair.
__device__ __forceinline__ void gemm2_sym(const float* __restrict__ A,
                                          const float* __restrict__ B,
                                          int ti, int tj0, int tj1,
                                          int lr, int kh,
                                          v8f& c0, v8f& c1) {
    v8f z = {0.f, 0.f, 0.f, 0.f, 0.f, 0.f, 0.f, 0.f};
    c0 = z; c1 = z;
    const float* arow  = &A[(ti  * 16 + lr) * PITCH];
    const float* brow0 = &B[(tj0 * 16 + lr) * PITCH];   // symmetric: row == column
    const float* brow1 = &B[(tj1 * 16 + lr) * PITCH];
    #pragma unroll
    for (int k0 = 0; k0 < 64; k0 += 4) {
        const int ka = k0 + 2 * kh;                     // even -> 8B aligned
        v2f a  = *(const v2f*)&arow [ka];
        v2f b0 = *(const v2f*)&brow0[ka];
        v2f b1 = *(const v2f*)&brow1[ka];
        c0 = __builtin_amdgcn_wmma_f32_16x16x4_f32(false, a, false, b0,
                                                   (short)0, c0, false, false);
        c1 = __builtin_amdgcn_wmma_f32_16x16x4_f32(false, a, false, b1,
                                                   (short)0, c1, false, false);
    }
}

__launch_bounds__(256)
__global__ void reeig_ns_kernel(const float* __restrict__ x,
                                float* __restrict__ out) {
    __shared__ float A0[64 * PITCH];   // (X - EP*I)/s  (kept for final |M| GEMM)
    __shared__ float Y [64 * PITCH];   // sign iterate
    __shared__ float T [64 * PITCH];   // Y*Y scratch
    __shared__ float rednorm;

    const int tid  = threadIdx.x;
    const int lane = tid & 31;
    const int wv   = tid >> 5;                 // 8 waves, 2 tiles each
    const int lr   = lane & 15;
    const int kh   = lane >> 4;                // 0/1
    const int ti   = wv >> 1;                  // shared row block of both tiles
    const int tj0  = (2 * wv) & 3;
    const int tj1  = tj0 + 1;
    const long base = (long)blockIdx.x * 4096;

    // ---- load M = X - EP*I into LDS, accumulate squared Frobenius norm ----
    if (tid == 0) rednorm = 0.f;
    __syncthreads();
    float part = 0.f;
    for (int i = tid; i < 4096; i += 256) {
        const int r = i >> 6, cc = i & 63;
        float v = x[base + i] - (r == cc ? EPV : 0.f);
        A0[r * PITCH + cc] = v;
        part += v * v;
    }
    #pragma unroll
    for (int off = 16; off; off >>= 1) part += __shfl_xor(part, off, 32);
    if (lane == 0) atomicAdd(&rednorm, part);
    __syncthreads();

    const float s  = sqrtf(rednorm) + 1e-30f;  // ||M||_2 <= ||M||_F = s
    const float rs = 1.f / s;

    // ---- Y0 = A0 = M/s ----
    for (int i = tid; i < 4096; i += 256) {
        const int r = i >> 6, cc = i & 63;
        float v = A0[r * PITCH + cc] * rs;
        A0[r * PITCH + cc] = v;
        Y [r * PITCH + cc] = v;
    }
    __syncthreads();

    // ---- Newton-Schulz: Y <- 1.5*Y - 0.5*Y*(Y*Y) ----
    for (int it = 0; it < NS_ITERS; ++it) {
        // T = Y * Y
        {
            v8f c0, c1;
            gemm2_sym(Y, Y, ti, tj0, tj1, lr, kh, c0, c1);
            const int mb = ti * 16 + kh * 8;
            const int n0 = tj0 * 16 + lr, n1 = tj1 * 16 + lr;
            #pragma unroll
            for (int r = 0; r < 8; ++r) {
                T[(mb + r) * PITCH + n0] = c0[r];
                T[(mb + r) * PITCH + n1] = c1[r];
            }
        }
        __syncthreads();
        // Z = Y * T  (kept in registers), then Y <- 1.5Y - 0.5Z
        v8f z0, z1;
        gemm2_sym(Y, T, ti, tj0, tj1, lr, kh, z0, z1);
        __syncthreads();                       // all reads of Y complete
        {
            const int mb = ti * 16 + kh * 8;
            const int n0 = tj0 * 16 + lr, n1 = tj1 * 16 + lr;
            #pragma unroll
            for (int r = 0; r < 8; ++r) {
                float y0 = Y[(mb + r) * PITCH + n0];
                float y1 = Y[(mb + r) * PITCH + n1];
                Y[(mb + r) * PITCH + n0] = 1.5f * y0 - 0.5f * z0[r];
                Y[(mb + r) * PITCH + n1] = 1.5f * y1 - 0.5f * z1[r];
            }
        }
        __syncthreads();
    }

    // ---- out = 0.5*(X + EP*I + s * (M/s)*sign(M)) , fused epilogue ----
    {
        v8f c0, c1;
        gemm2_sym(A0, Y, ti, tj0, tj1, lr, kh, c0, c1);
        const int mb = ti * 16 + kh * 8;
        const int n0 = tj0 * 16 + lr, n1 = tj1 * 16 + lr;
        #pragma unroll
        for (int r = 0; r < 8; ++r) {
            const int m = mb + r;
            const float x0 = x[base + m * 64 + n0];
            const float x1 = x[base + m * 64 + n1];
            out[base + m * 64 + n0] = 0.5f * (x0 + (m == n0 ? EPV : 0.f) + s * c0[r]);
            out[base + m * 64 + n1] = 0.5f * (x1 + (m == n1 ? EPV : 0.f) + s * c1[r]);
        }
    }
}

extern "C" void kernel_launch(void* const* d_in, const int* in_sizes, int n_in,
                              void* d_out, int out_size, void* d_ws, size_t ws_size,
                              hipStream_t stream) {
    const float* x = (const float*)d_in[0];
    float* out = (float*)d_out;
    const int batches = in_sizes[0] / 4096;    // 8192 matrices of 64x64
    reeig_ns_kernel<<<batches, 256, 0, stream>>>(x, out);
}